// GSNN_15401752723587
// MI455X (gfx1250) — compile-verified
//
#include <hip/hip_runtime.h>
#include <hip/hip_bf16.h>

// ---------------------------------------------------------------------------
// GSNN: 10 layers of per-node tiny GEMMs on MI455X (gfx1250, wave32, WMMA).
//
// Per node f:  h = ELU( g(64x8) @ W1^T(8x16) + b1 ),  o = h @ W2^T(16x8) + b2
// Mapped to V_WMMA_F32_16X16X4_F32 chains (K=8 -> 2 steps, K=16 -> 4 steps),
// 4 M-tiles cover the batch of 64.  One wave per node, 8 waves per block.
// Working set (~82 MB) is L2-resident (192 MB) -> L2-BW bound; fp32 WMMA
// keeps reference precision and wastes nothing on K for these tiny shapes.
// ---------------------------------------------------------------------------

typedef __attribute__((ext_vector_type(2))) float v2f;
typedef __attribute__((ext_vector_type(8))) float v8f;

#define F_NODES 16384
#define DEG 8
#define EDGES (F_NODES * DEG)   // 131072
#define BATCH 64
#define HID 16
#define LAYERS 10
#define WAVES_PER_BLOCK 8
#define LDS_STRIDE 18           // padded 16x16 f32 tile row stride (banks)

__device__ __forceinline__ float elu_f(float z) {
    return z > 0.0f ? z : (__expf(z) - 1.0f);
}

__global__ __launch_bounds__(256)
void gsnn_layer(const float* __restrict__ xin,   // (B, E) current activations
                const float* __restrict__ x0,    // (B, E) residual source
                const float* __restrict__ W1,    // (F, H, D)
                const float* __restrict__ b1,    // (F, H)
                const float* __restrict__ W2,    // (F, D, H)
                const float* __restrict__ b2,    // (F, D)
                const int*   __restrict__ ixs,   // (F, D) gather columns
                float*       __restrict__ xout)  // (B, E)
{
    __shared__ float lds_h[WAVES_PER_BLOCK * 16 * LDS_STRIDE];

    const int tid  = threadIdx.x;
    const int wave = tid >> 5;
    const int lane = tid & 31;
    const int n    = lane & 15;   // N index (H dim / batch row for A loads)
    const int hi   = lane >> 4;   // half-wave selector (K pairing)

    const int f = blockIdx.x * WAVES_PER_BLOCK + wave;   // node id
    float* hbuf = &lds_h[wave * 16 * LDS_STRIDE];

    // ---- per-node constant operands -------------------------------------
    // B-operand of GEMM1: B[k][n] = W1[f][n][k].  Lane supplies k = k0+2*hi
    // and k0+2*hi+1 (adjacent in memory -> float2).
    const float* W1f = W1 + (size_t)f * (HID * DEG);
    const v2f wb1_0 = *(const v2f*)(W1f + n * DEG + 2 * hi);       // k0 = 0
    const v2f wb1_1 = *(const v2f*)(W1f + n * DEG + 4 + 2 * hi);   // k0 = 4
    const float bias1 = b1[f * HID + n];

    // B-operand of GEMM2: B[k][n] = W2[f][n][k], N padded 8 -> 16 with zeros.
    const float* W2f = W2 + (size_t)f * (DEG * HID);
    v2f wb2[4];
    float bias2 = 0.0f;
    if (n < DEG) {
        #pragma unroll
        for (int j = 0; j < 4; ++j)
            wb2[j] = *(const v2f*)(W2f + n * HID + 4 * j + 2 * hi);
        bias2 = b2[f * DEG + n];
    } else {
        #pragma unroll
        for (int j = 0; j < 4; ++j) { wb2[j].x = 0.0f; wb2[j].y = 0.0f; }
    }

    // Gather columns this lane needs for the A-operand of GEMM1.
    const int* ixf = ixs + f * DEG;
    const int c00 = ixf[2 * hi];
    const int c01 = ixf[2 * hi + 1];
    const int c10 = ixf[4 + 2 * hi];
    const int c11 = ixf[4 + 2 * hi + 1];

    const size_t ocol = (size_t)f * DEG + n;   // output column (valid if n<8)

    // ---- 4 M-tiles of 16 batch rows -------------------------------------
    #pragma unroll
    for (int mt = 0; mt < 4; ++mt) {
        const int mb = mt * 16;

        // A-operand of GEMM1: lane row M=n, K pair per half-wave (gather).
        const float* xrow = xin + (size_t)(mb + n) * EDGES;
        v2f a0; a0.x = xrow[c00]; a0.y = xrow[c01];
        v2f a1; a1.x = xrow[c10]; a1.y = xrow[c11];

        // GEMM1: (16x8) x (8x16) via two chained f32 WMMAs.
        v8f acc = {};
        acc = __builtin_amdgcn_wmma_f32_16x16x4_f32(
                  false, a0, false, wb1_0, (short)0, acc, false, false);
        acc = __builtin_amdgcn_wmma_f32_16x16x4_f32(
                  false, a1, false, wb1_1, (short)0, acc, false, false);

        // bias + ELU, spill h tile to LDS in [m][k] order (D->A transpose).
        #pragma unroll
        for (int r = 0; r < 8; ++r) {
            float hv = elu_f(acc[r] + bias1);
            hbuf[(r + 8 * hi) * LDS_STRIDE + n] = hv;
        }
        __syncthreads();   // uniform across block (same trip count all waves)

        // A-operand of GEMM2 from LDS: lane row M=n, K pairs k0+2*hi.
        v2f a2[4];
        #pragma unroll
        for (int j = 0; j < 4; ++j)
            a2[j] = *(const v2f*)(hbuf + n * LDS_STRIDE + 4 * j + 2 * hi);

        // GEMM2: (16x16) x (16x16-padded) via four chained f32 WMMAs.
        v8f o = {};
        #pragma unroll
        for (int j = 0; j < 4; ++j)
            o = __builtin_amdgcn_wmma_f32_16x16x4_f32(
                    false, a2[j], false, wb2[j], (short)0, o, false, false);

        __syncthreads();   // protect LDS tile before next M-tile overwrites

        // Epilogue: o + b2 + x0 residual; only N columns 0..7 are real.
        if (n < DEG) {
            #pragma unroll
            for (int r = 0; r < 8; ++r) {
                const size_t off = (size_t)(mb + r + 8 * hi) * EDGES + ocol;
                xout[off] = o[r] + bias2 + x0[off];
            }
        }
    }
}

extern "C" void kernel_launch(void* const* d_in, const int* in_sizes, int n_in,
                              void* d_out, int out_size, void* d_ws, size_t ws_size,
                              hipStream_t stream) {
    (void)in_sizes; (void)n_in; (void)out_size; (void)ws_size;

    const float* x0 = (const float*)d_in[0];   // (B, E)
    const float* W1 = (const float*)d_in[1];   // (F, H, D)
    const float* b1 = (const float*)d_in[2];   // (F, H)
    const float* W2 = (const float*)d_in[3];   // (F, D, H)
    const float* b2 = (const float*)d_in[4];   // (F, D)
    const int*   ix = (const int*)d_in[5];     // (F, D)

    float* bufA = (float*)d_ws;    // 32 MB ping buffer (even-layer outputs)
    float* out  = (float*)d_out;   // doubles as pong buffer; layer 9 -> here

    const dim3 grid(F_NODES / WAVES_PER_BLOCK);   // 2048 blocks
    const dim3 block(WAVES_PER_BLOCK * 32);       // 256 threads = 8 waves

    const float* cur = x0;
    for (int l = 0; l < LAYERS; ++l) {
        float* dst = (l & 1) ? out : bufA;        // odd layers -> d_out
        gsnn_layer<<<grid, block, 0, stream>>>(cur, x0, W1, b1, W2, b2, ix, dst);
        cur = dst;
    }
}